// GNN_10514079941483
// MI455X (gfx1250) — compile-verified
//
#include <hip/hip_runtime.h>
#include <hip/hip_bf16.h>
#include <stdint.h>

typedef __attribute__((ext_vector_type(2))) float v2f;
typedef __attribute__((ext_vector_type(8))) float v8f;

static constexpr int NN   = 50000;  // nodes
static constexpr int INC  = 512;
static constexpr int HID  = 256;
static constexpr int OUTC = 128;

// ---------------------------------------------------------------- utilities
__device__ inline unsigned ordf(float f) {
    unsigned u = __float_as_uint(f);
    return (u & 0x80000000u) ? ~u : (u | 0x80000000u);
}
__device__ inline float deordf(unsigned u) {
    return (u & 0x80000000u) ? __uint_as_float(u ^ 0x80000000u)
                             : __uint_as_float(~u);
}

// ---------------------------------------------------------------- degree
__global__ void fill_ones_int(int* p, int n) {
    int i = blockIdx.x * blockDim.x + threadIdx.x;
    if (i < n) p[i] = 1;                       // self-loop contributes 1
}
__global__ void init_small(unsigned* maxord, int* minidx, int F) {
    int c = threadIdx.x;
    if (c < F) { maxord[c] = 0u; minidx[c] = 0x7fffffff; }
}
__global__ void deg_count(const int* __restrict__ dst, int* __restrict__ deg, int E) {
    int e = blockIdx.x * blockDim.x + threadIdx.x;
    if (e < E) atomicAdd(&deg[dst[e]], 1);
}
__global__ void deg_to_dinv(float* buf, int n) {   // in-place int -> rsqrt(float)
    int i = blockIdx.x * blockDim.x + threadIdx.x;
    if (i < n) {
        int v = ((const int*)buf)[i];
        buf[i] = rsqrtf((float)v);             // deg >= 1 always (self-loop)
    }
}

// ---------------------------------------------------------------- fp32 WMMA GEMM
// C[M,N] = A[M,K]*B[K,N] (+bias). Workgroup = 8 waves sharing one 32-column
// strip of B staged through LDS in 256-row K-chunks. LDS layout is
// "pair-interleaved + XOR-32 swizzle": element (kr, c) lives at
//   (kr/2)*64 + ((2c + ((kr/2)&1)*32) & 63) + (kr&1)
// so each lane's B fragment is ONE aligned ds_load_b64 (v2f lands directly in
// the VGPR pair WMMA wants -> no marshalling moves), and the two half-waves
// (even/odd pair-rows) hit disjoint bank halves 0-31 / 32-63 -> conflict-free.
template <int K, int N>
__global__ __launch_bounds__(256)
void gemm_wmma_lds(const float* __restrict__ A, const float* __restrict__ B,
                   const float* __restrict__ bias, float* __restrict__ C, int M) {
    constexpr int NT2 = N / 32;                // 16-col tile pairs
    constexpr int KC  = (K > 256) ? 256 : K;   // K-chunk staged in LDS (<= 32KB)
    __shared__ __align__(16) float Bs[KC * 32];

    const int Mtiles = M >> 4;
    const int np  = blockIdx.x % NT2;          // which 32-column strip
    const int mb  = blockIdx.x / NT2;          // block of 8 M-tiles
    const int tid = threadIdx.x;

    const int waveId = tid >> 5;
    const int mt     = mb * 8 + waveId;
    const bool active = (mt < Mtiles);
    const int lane = tid & 31;
    const int l16  = lane & 15;
    const int hi   = lane >> 4;                // 0 -> K{0,1}, 1 -> K{2,3}
    const int koff = hi << 1;

    const float* Arow = active ? (A + (size_t)(mt * 16 + l16) * K + koff)
                               : (A + (size_t)l16 * K + koff);
    if (active) __builtin_prefetch(Arow, 0, 3);    // global_prefetch_b8

    // per-lane LDS read bases (swizzle folded in); k-loop adds k*32 floats
    const float* base0 = Bs + hi * 64 + ((2 * l16      + (hi << 5)) & 63);
    const float* base1 = Bs + hi * 64 + ((2 * l16 + 32 + (hi << 5)) & 63);

    v8f acc0 = {}; v8f acc1 = {};
    for (int k0 = 0; k0 < K; k0 += KC) {
        __syncthreads();                       // previous chunk fully consumed
        // cooperative stage: B[k0..k0+KC, np*32..+32] -> swizzled LDS
        for (int i = tid; i < KC * 32; i += 256) {
            int kr = i >> 5, c = i & 31;
            int kp = kr >> 1;
            int idx = kp * 64 + ((2 * c + ((kp & 1) << 5)) & 63) + (kr & 1);
            Bs[idx] = B[(size_t)(k0 + kr) * N + np * 32 + c];
        }
        __syncthreads();
        if (active) {
            #pragma unroll 4
            for (int k = 0; k < KC; k += 4) {
                v2f a  = *(const v2f*)(Arow + k0 + k);     // global, 8B aligned
                v2f b0 = *(const v2f*)(base0 + (k << 5));  // ds_load_b64
                v2f b1 = *(const v2f*)(base1 + (k << 5));  // ds_load_b64
                acc0 = __builtin_amdgcn_wmma_f32_16x16x4_f32(
                           false, a, false, b0, (short)0, acc0, false, false);
                acc1 = __builtin_amdgcn_wmma_f32_16x16x4_f32(
                           false, a, false, b1, (short)0, acc1, false, false);
            }
        }
    }
    if (!active) return;
    // C/D layout: VGPR r -> row mt*16 + r (+8 for upper half-wave), col = lane&15
    int rbase = mt * 16 + (hi << 3);
    int c0    = np * 32 + l16;
    float bi0 = bias ? bias[c0]      : 0.f;
    float bi1 = bias ? bias[c0 + 16] : 0.f;
    #pragma unroll
    for (int r = 0; r < 8; ++r) {
        size_t row = (size_t)(rbase + r);
        C[row * N + c0]      = acc0[r] + bi0;
        C[row * N + c0 + 16] = acc1[r] + bi1;
    }
}

// ---------------------------------------------------------------- aggregation
__global__ void selfloop_init(const float* __restrict__ hl, const float* __restrict__ dinv,
                              float* __restrict__ agg, int F, int total) {
    int idx = blockIdx.x * blockDim.x + threadIdx.x;
    if (idx >= total) return;
    int node = idx / F;
    float d = dinv[node];
    agg[idx] = d * d * hl[idx];
}

// one wave per edge; 32 lanes stripe the F channels
__global__ __launch_bounds__(256)
void edge_scatter(const int* __restrict__ src, const int* __restrict__ dst,
                  const float* __restrict__ dinv, const float* __restrict__ hl,
                  float* __restrict__ agg, int F, int E) {
    int wave = (blockIdx.x * blockDim.x + threadIdx.x) >> 5;
    int lane = threadIdx.x & 31;
    if (wave >= E) return;
    int s = src[wave], d = dst[wave];
    float w = dinv[s] * dinv[d];
    const float* hs = hl + (size_t)s * F;
    float* ad = agg + (size_t)d * F;
    for (int c = lane; c < F; c += 32)
        atomicAdd(&ad[c], w * hs[c]);          // global_atomic_add_f32; L2-resident
}

__global__ void bias_act(const float* __restrict__ agg, const float* __restrict__ bias,
                         float* __restrict__ out, int F, int relu, int total) {
    int idx = blockIdx.x * blockDim.x + threadIdx.x;
    if (idx >= total) return;
    float v = agg[idx] + bias[idx % F];
    out[idx] = relu ? fmaxf(v, 0.f) : v;
}

// ---------------------------------------------------------------- column max / argmax
__global__ __launch_bounds__(256)
void colmax_pass1(const float* __restrict__ p, unsigned* __restrict__ maxord, int M, int F) {
    int gtid   = blockIdx.x * blockDim.x + threadIdx.x;
    int total  = gridDim.x * blockDim.x;
    int c      = gtid & (F - 1);
    int stride = total >> 7;                   // F == 128
    unsigned m = 0;
    for (int r = gtid >> 7; r < M; r += stride)
        m = max(m, ordf(p[(size_t)r * F + c]));
    atomicMax(&maxord[c], m);
}
__global__ __launch_bounds__(256)
void argmax_pass2(const float* __restrict__ p, const unsigned* __restrict__ maxord,
                  int* __restrict__ minidx, int M, int F) {
    int gtid   = blockIdx.x * blockDim.x + threadIdx.x;
    int total  = gridDim.x * blockDim.x;
    int c      = gtid & (F - 1);
    int stride = total >> 7;
    unsigned target = maxord[c];
    int best = 0x7fffffff;
    for (int r = gtid >> 7; r < M; r += stride)
        if (ordf(p[(size_t)r * F + c]) == target) { best = r; break; } // rows ascending
    if (best != 0x7fffffff) atomicMin(&minidx[c], best);
}
__global__ void finalize_out(const unsigned* maxord, const int* minidx,
                             float* vals, float* idxf, int F) {
    int c = threadIdx.x;
    if (c < F) {
        vals[c] = deordf(maxord[c]);
        idxf[c] = (float)minidx[c];            // output buffer is float
    }
}

// ---------------------------------------------------------------- launch
extern "C" void kernel_launch(void* const* d_in, const int* in_sizes, int n_in,
                              void* d_out, int out_size, void* d_ws, size_t ws_size,
                              hipStream_t stream) {
    const float* x  = (const float*)d_in[0];
    const int*   ei = (const int*)d_in[1];     // [2, E] : row0=src, row1=dst
    const float* W0 = (const float*)d_in[2];
    const float* b0 = (const float*)d_in[3];
    const float* W1 = (const float*)d_in[4];
    const float* b1 = (const float*)d_in[5];
    const float* PW = (const float*)d_in[6];
    const float* Pb = (const float*)d_in[7];
    const int E = in_sizes[1] / 2;
    const int* esrc = ei;
    const int* edst = ei + E;

    // workspace carve-up (256B aligned)
    size_t off = 0;
    auto carve = [&](size_t bytes) { size_t o = off; off += (bytes + 255) & ~(size_t)255; return o; };
    char* ws = (char*)d_ws;
    float*    dinv   = (float*)   (ws + carve((size_t)NN * 4));
    float*    bufA   = (float*)   (ws + carve((size_t)NN * HID * 4));  // hl1 / h1 / p
    float*    bufB   = (float*)   (ws + carve((size_t)NN * HID * 4));  // agg1 / hl2
    unsigned* maxord = (unsigned*)(ws + carve(OUTC * 4));
    int*      minidx = (int*)     (ws + carve(OUTC * 4));

    float* h    = (float*)d_out;               // [NN, OUTC]
    float* vals = h + (size_t)NN * OUTC;       // [OUTC]
    float* idxf = vals + OUTC;                 // [OUTC]

    const int Mtiles  = NN / 16;               // 3125
    const int mblocks = (Mtiles + 7) / 8;      // 391

    // ---- degree / normalization
    fill_ones_int<<<(NN + 255) / 256, 256, 0, stream>>>((int*)dinv, NN);
    init_small<<<1, 128, 0, stream>>>(maxord, minidx, OUTC);
    deg_count<<<(E + 255) / 256, 256, 0, stream>>>(edst, (int*)dinv, E);
    deg_to_dinv<<<(NN + 255) / 256, 256, 0, stream>>>(dinv, NN);

    // ---- layer 1: h1 = relu(Agg(x@W0) + b0)
    {
        gemm_wmma_lds<INC, HID><<<mblocks * (HID / 32), 256, 0, stream>>>(x, W0, nullptr, bufA, NN);
        int tot = NN * HID;
        selfloop_init<<<(tot + 255) / 256, 256, 0, stream>>>(bufA, dinv, bufB, HID, tot);
        edge_scatter<<<(E * 32 + 255) / 256, 256, 0, stream>>>(esrc, edst, dinv, bufA, bufB, HID, E);
        bias_act<<<(tot + 255) / 256, 256, 0, stream>>>(bufB, b0, bufA, HID, 1, tot);
    }
    // ---- layer 2: h = Agg(h1@W1) + b1  (written to d_out)
    {
        gemm_wmma_lds<HID, OUTC><<<mblocks * (OUTC / 32), 256, 0, stream>>>(bufA, W1, nullptr, bufB, NN);
        int tot = NN * OUTC;
        selfloop_init<<<(tot + 255) / 256, 256, 0, stream>>>(bufB, dinv, h, OUTC, tot);
        edge_scatter<<<(E * 32 + 255) / 256, 256, 0, stream>>>(esrc, edst, dinv, bufB, h, OUTC, E);
        bias_act<<<(tot + 255) / 256, 256, 0, stream>>>(h, b1, h, OUTC, 0, tot);
    }
    // ---- head: p = h@PW + Pb ; column max + first-argmax
    {
        gemm_wmma_lds<OUTC, OUTC><<<mblocks * (OUTC / 32), 256, 0, stream>>>(h, PW, Pb, bufA, NN);
        colmax_pass1<<<256, 256, 0, stream>>>(bufA, maxord, NN, OUTC);
        argmax_pass2<<<256, 256, 0, stream>>>(bufA, maxord, minidx, NN, OUTC);
        finalize_out<<<1, 128, 0, stream>>>(maxord, minidx, vals, idxf, OUTC);
    }
}